// CenterHeadTemplate_31490700214819
// MI455X (gfx1250) — compile-verified
//
#include <hip/hip_runtime.h>

typedef __attribute__((ext_vector_type(2))) float v2f;
typedef __attribute__((ext_vector_type(8))) float v8f;
typedef __attribute__((ext_vector_type(4))) unsigned int v4u;
typedef __attribute__((ext_vector_type(8))) int v8i;
typedef __attribute__((ext_vector_type(4))) int v4i;

#define FEAT 256
#define MAXN 64
#define NHOT 49

#if defined(__has_builtin)
#if __has_builtin(__builtin_amdgcn_tensor_store_from_lds) && \
    __has_builtin(__builtin_amdgcn_s_wait_tensorcnt)
#define HAVE_TDM 1
#endif
#endif

// One block per box. 128 threads = 4 wave32s.
// Phase 0: TDM-zero this box's 256KB heat tile (one tensor_store_from_lds,
//          64 iterations of a 4KB zero LDS tile); VMEM-zero reg/mask/ind.
// Phase 3 (overlapped with TDM): hot-point compaction -> reg / ind / mask.
// Phase 1: rotate FG_OFF grid via V_WMMA_F32_16X16X4_F32, scatter rsq(d2).
// Phase 2: fake points (0.8) + center (1.0).
__global__ __launch_bounds__(128)
void centerhead_assign(const float* __restrict__ gt,
                       const float* __restrict__ hgt,
                       float* __restrict__ heat,
                       float* __restrict__ reg,
                       float* __restrict__ maskf,
                       float* __restrict__ indf) {
  const int box  = blockIdx.x;
  const int tid  = threadIdx.x;
  const int lane = tid & 31;
  const int wave = tid >> 5;

  const float* bb = gt + (size_t)box * 8;
  __builtin_prefetch(bb, 0, 1);  // global_prefetch_b8

  const float bx = bb[0], by = bb[1];
  const float d0 = bb[3], d1 = bb[4], d2 = bb[5];
  const float ryaw = bb[6];
  const float h = hgt[box];

  // match reference's division chain exactly
  const float ctf_x = (bx + 51.2f) / 0.1f / 4.0f;
  const float ctf_y = (by + 51.2f) / 0.1f / 4.0f;
  const int cti_x = (int)truncf(ctf_x);
  const int cti_y = (int)truncf(ctf_y);
  const int w2d = (int)truncf(d0 / 0.1f / 8.0f);
  const int l2d = (int)truncf(d1 / 0.1f / 8.0f);
  const float c = cosf(ryaw), s = sinf(ryaw);

  float* hmap = heat + (size_t)box * FEAT * FEAT;

#ifdef HAVE_TDM
  __shared__ __align__(16) float zbuf[1024];  // 4KB zero tile for the TDM
#endif
  __shared__ int vflag[NHOT];
  __shared__ int pxx[NHOT];
  __shared__ int pyy[NHOT];

  // ---- Phase 0: zeros ----
  {
    float4 z4 = make_float4(0.f, 0.f, 0.f, 0.f);
#ifdef HAVE_TDM
    {
      float4* zb4 = (float4*)zbuf;
      zb4[tid] = z4;          // 1024 floats = 256 f4; 128 threads x 2
      zb4[tid + 128] = z4;
    }
#else
    float4* h4 = (float4*)hmap;
    #pragma unroll 4
    for (int i = tid; i < FEAT * FEAT / 4; i += 128) h4[i] = z4;
#endif
    float4* r4 = (float4*)(reg + (size_t)box * MAXN * 8);  // 512 floats
    r4[tid] = z4;
    if (tid < MAXN) {
      maskf[(size_t)box * MAXN + tid] = 0.0f;
      indf[(size_t)box * MAXN + tid] = 0.0f;
    }
  }
  // hot-point flags (LDS) for the compaction scan
  if (tid < NHOT) {
    const int X = cti_x + (tid / 7 - 3);
    const int Y = cti_y + (tid % 7 - 3);
    pxx[tid] = X;
    pyy[tid] = Y;
    vflag[tid] = (X >= 0 && X < FEAT && Y >= 0 && Y < FEAT) ? 1 : 0;
  }
  __threadfence();
  __syncthreads();  // LDS zero tile + reg zeros complete

#ifdef HAVE_TDM
  // ---- Issue TDM: store 4KB zero tile 64x -> 256KB heat tile ----
  if (tid < 32) {  // wave 0, full EXEC
    const unsigned long long gaddr = (unsigned long long)(uintptr_t)hmap;
    const unsigned int lds_off = (unsigned int)((uintptr_t)zbuf & 0xffffffffu);
    v4u g0;
    g0[0] = 1u;                                  // count=1 (valid user D#)
    g0[1] = lds_off;                             // lds_addr (bytes)
    g0[2] = (unsigned int)gaddr;                 // global_addr[31:0]
    g0[3] = (unsigned int)((gaddr >> 32) & 0x1FFFFFFull) | (2u << 30);  // [56:32] + type=2
    v8i g1;
    g1[0] = (2 << 16) | (1 << 19);               // data_size=4B, iterate_enable=1
    g1[1] = (int)(1024u << 16);                  // tensor_dim0[15:0] = 1024
    g1[2] = (int)(1u << 16);                     // tensor_dim1[15:0] = 1
    g1[3] = (int)(1024u << 16);                  // tile_dim0 = 1024
    g1[4] = 1;                                   // tile_dim1 = 1 (tile_dim2 = 0)
    g1[5] = 1024;                                // tensor_dim0_stride = 1024
    g1[6] = 0;
    g1[7] = 0;
    v4i g2;
    g2[0] = 1;                                   // tensor_dim2 (unused)
    g2[1] = 0;                                   // lds_addr_increment = 0 (reuse tile)
    g2[2] = 1024;                                // global_addr_increment (elements)
    g2[3] = (63 << 16);                          // iterate_count = 63 -> 64 iterations
    v4i g3 = {0, 0, 0, 0};
    v8i g4 = {0, 0, 0, 0, 0, 0, 0, 0};           // unused trailing group (VADDR4)
    __builtin_amdgcn_tensor_store_from_lds(g0, g1, g2, g3, g4, 0);
  }
#endif

  // ---- Phase 3 (overlaps TDM): hot points -> reg / ind / mask ----
  if (tid < NHOT && vflag[tid]) {
    int pos = 0;
    for (int k = 0; k < tid; ++k) pos += vflag[k];  // stable: valid first, original order
    float* r = reg + ((size_t)box * MAXN + pos) * 8;
    r[0] = ctf_x - (float)pxx[tid];
    r[1] = ctf_y - (float)pyy[tid];
    r[2] = h;
    r[3] = d0;
    r[4] = d1;
    r[5] = d2;
    r[6] = s;  // sin(ry)
    r[7] = c;  // cos(ry)
    indf[(size_t)box * MAXN + pos]  = (float)(pyy[tid] * FEAT + pxx[tid]);
    maskf[(size_t)box * MAXN + pos] = 1.0f;
  }

#ifdef HAVE_TDM
  if (tid < 32) __builtin_amdgcn_s_wait_tensorcnt(0);  // heat tile zeroed
#endif
  __threadfence();
  __syncthreads();

  // ---- Phase 1: FG scatter via WMMA rotation ----
  // A (16x4 f32): row0 = [c, -s, 0, 0], row1 = [s, c, 0, 0], rows 2..15 = 0.
  v2f A;
  A.x = (lane == 0) ? c : ((lane == 1) ? s : 0.0f);
  A.y = (lane == 0) ? -s : ((lane == 1) ? c : 0.0f);

  const float extw = (float)(w2d + 1);
  const float extl = (float)(l2d + 1);

  // 729 FG points, 16 per WMMA -> 46 groups, round-robined over 4 waves.
  for (int g = wave; g < 46; g += 4) {
    const int p = g * 16 + (lane & 15);  // point index (column N = lane, lanes 0-15)
    const float off0 = (float)(p / 27 - 13);
    const float off1 = (float)(p % 27 - 13);
    // B (4x16 f32): lanes 0-15 hold rows K=0 (x) / K=1 (y); lanes 16-31 K=2/3 = 0
    v2f Bv;
    Bv.x = (lane < 16) ? off0 : 0.0f;
    Bv.y = (lane < 16) ? off1 : 0.0f;
    v8f acc = {};
    acc = __builtin_amdgcn_wmma_f32_16x16x4_f32(
        /*neg_a=*/false, A, /*neg_b=*/false, Bv,
        /*c_mod=*/(short)0, acc, /*reuse_a=*/false, /*reuse_b=*/false);
    const float rx = acc[0];  // D[0][n] on lane n
    const float ry = acc[1];  // D[1][n] on lane n
    if (lane < 16 && p < 729) {
      const bool ext_ok = (fabsf(off0) <= extw) && (fabsf(off1) <= extl);
      const float tx = truncf(rx);
      const float ty = truncf(ry);
      const int fx = (int)tx + cti_x;
      const int fy = (int)ty + cti_y;
      if (ext_ok && fx >= 0 && fx < FEAT && fy >= 0 && fy < FEAT) {
        hmap[fy * FEAT + fx] = __frsqrt_rn(tx * tx + ty * ty);  // v_rsq_f32
      }
    }
  }
  __threadfence();
  __syncthreads();

  // ---- Phase 2: fake points then center (overwrite FG values) ----
  if (tid < 4) {
    const int fox[4] = {1, -1, 0, 0};
    const int foy[4] = {0, 0, 1, -1};
    const int fx = cti_x + fox[tid];
    const int fy = cti_y + foy[tid];
    if (fx >= 0 && fx < FEAT && fy >= 0 && fy < FEAT) hmap[fy * FEAT + fx] = 0.8f;
  } else if (tid == 4) {
    if (cti_x >= 0 && cti_x < FEAT && cti_y >= 0 && cti_y < FEAT)
      hmap[cti_y * FEAT + cti_x] = 1.0f;
  }
}

extern "C" void kernel_launch(void* const* d_in, const int* in_sizes, int n_in,
                              void* d_out, int out_size, void* d_ws, size_t ws_size,
                              hipStream_t stream) {
  (void)n_in; (void)d_ws; (void)ws_size; (void)out_size;
  const float* gt  = (const float*)d_in[0];   // (B,1,8) f32
  const float* hgt = (const float*)d_in[1];   // (B,)    f32
  const int B = in_sizes[1];

  float* out   = (float*)d_out;
  float* heat  = out;                                   // B*256*256
  float* reg   = heat + (size_t)B * FEAT * FEAT;        // B*64*8
  float* maskf = reg + (size_t)B * MAXN * 8;            // B*64
  float* indf  = maskf + (size_t)B * MAXN;              // B*64

  centerhead_assign<<<B, 128, 0, stream>>>(gt, hgt, heat, reg, maskf, indf);
}